// GCN_80041010528430
// MI455X (gfx1250) — compile-verified
//
#include <hip/hip_runtime.h>
#include <hip/hip_bf16.h>

typedef __attribute__((ext_vector_type(2))) float v2f;
typedef __attribute__((ext_vector_type(8))) float v8f;

#define D_IN  512
#define D_H   128
#define D_OUT 32

// ---------------------------------------------------------------------------
// Degree / normalization kernels
// ---------------------------------------------------------------------------
__global__ void init_deg_kernel(float* __restrict__ deg, int n) {
    int i = blockIdx.x * blockDim.x + threadIdx.x;
    if (i < n) deg[i] = 1.0f;  // self loop
}

__global__ void edge_deg_kernel(const int* __restrict__ dst, float* __restrict__ deg, int e) {
    int i = blockIdx.x * blockDim.x + threadIdx.x;
    if (i < e) atomicAdd(&deg[dst[i]], 1.0f);
}

__global__ void dinv_kernel(const float* __restrict__ deg, float* __restrict__ dinv, int n) {
    int i = blockIdx.x * blockDim.x + threadIdx.x;
    if (i < n) dinv[i] = rsqrtf(fmaxf(deg[i], 1.0f));
}

// ---------------------------------------------------------------------------
// WMMA GEMM: C[M x N] = A[M x K] @ W[K x N] (+bias)(+relu)
// One wave owns a 16-row strip and all N columns (NT = N/16 accumulator tiles).
// W is staged through LDS in K-PAIR-INTERLEAVED layout:
//   lds[p*N + c] = { W[k0+2p][c], W[k0+2p+1][c] }
// so each WMMA B fragment is a single aligned ds_load_b64 that lands directly
// in the even-aligned VGPR pair the matrix op consumes (no repack moves).
// A fragment = one float2 global load per lane (16x4 f32 A layout puts the
// lane's K-pair at k0+kk+2*(lane>>4), contiguous in memory).
// fp32 WMMA (V_WMMA_F32_16X16X4_F32) -> bit-exact fp32 math.
// ---------------------------------------------------------------------------
template <int K, int N, int KC, bool BIAS, bool RELU>
__global__ __launch_bounds__(256) void gemm_wmma_kernel(
    const float* __restrict__ A, const float* __restrict__ W,
    const float* __restrict__ bias, float* __restrict__ C, int M) {
    constexpr int NT = N / 16;
    constexpr int KP = KC / 2;                       // K-pairs per chunk
    __shared__ float2 lds[KP * N];

    const int tid  = threadIdx.x;
    const int lane = tid & 31;
    const int wave = tid >> 5;
    const int tile = blockIdx.x * 8 + wave;          // 16-row strip index
    const bool active = (tile * 16) < M;             // wave-uniform -> EXEC all 1s inside
    const int rowA = lane & 15;
    const int col  = lane & 15;
    const int kh2  = lane >> 4;                      // which K-pair of the frag (0/1)
    const long arow = (long)(tile * 16 + rowA) * K;

    v8f acc[NT];
#pragma unroll
    for (int t = 0; t < NT; ++t)
        acc[t] = (v8f){0.f, 0.f, 0.f, 0.f, 0.f, 0.f, 0.f, 0.f};

    for (int k0 = 0; k0 < K; k0 += KC) {
        // cooperative fill: pair-interleave rows 2p and 2p+1 of the W chunk
        for (int i = tid; i < KP * N; i += 256) {
            const int p = i / N;
            const int c = i - p * N;
            const long g = (long)(k0 + 2 * p) * N + c;
            lds[i] = make_float2(W[g], W[g + N]);
        }
        __syncthreads();

        if (active) {
            for (int kk2 = 0; kk2 < KP; kk2 += 2) {  // 4 K values per iteration
                v2f a = *(const v2f*)(A + arow + k0 + 2 * (kk2 + kh2));
                const float2* bp = lds + (kk2 + kh2) * N + col;
#pragma unroll
                for (int t = 0; t < NT; ++t) {
                    v2f b = *(const v2f*)(bp + t * 16);   // single ds_load_b64
                    acc[t] = __builtin_amdgcn_wmma_f32_16x16x4_f32(
                        false, a, false, b, (short)0, acc[t], false, false);
                }
            }
        }
        __syncthreads();
    }

    if (active) {
        const int mh = (lane >> 4) << 3;             // 0 or 8 (C/D row half)
#pragma unroll
        for (int t = 0; t < NT; ++t) {
            const int c = t * 16 + col;
            const float bv = BIAS ? bias[c] : 0.f;
#pragma unroll
            for (int v = 0; v < 8; ++v) {
                float val = acc[t][v] + bv;
                if (RELU) val = fmaxf(val, 0.f);
                C[(long)(tile * 16 + mh + v) * N + c] = val;
            }
        }
    }
}

// ---------------------------------------------------------------------------
// Message passing (D_H = 128 features). Self-loop init, then wave-per-edge
// gather + scatter-add. h (51 MB) stays resident in the 192 MB L2, so both the
// random gathers and the float atomics run at L2 bandwidth.
// ---------------------------------------------------------------------------
__global__ void selfloop_kernel(const float* __restrict__ h,
                                const float* __restrict__ dinv,
                                float* __restrict__ agg, int n) {
    long idx = (long)blockIdx.x * blockDim.x + threadIdx.x;
    if (idx >= (long)n * D_H) return;
    int i = (int)(idx >> 7);
    float di = dinv[i];
    agg[idx] = h[idx] * di * di;
}

__global__ void edge_scatter_kernel(const float* __restrict__ h,
                                    const int* __restrict__ src,
                                    const int* __restrict__ dst,
                                    const float* __restrict__ dinv,
                                    float* __restrict__ agg, int e) {
    const int lane = threadIdx.x & 31;
    int w      = (int)(((long)blockIdx.x * blockDim.x + threadIdx.x) >> 5);
    int nwaves = (int)(((long)gridDim.x * blockDim.x) >> 5);
    for (int ed = w; ed < e; ed += nwaves) {
        const int s = src[ed];
        const int d = dst[ed];
        const float nrm = dinv[s] * dinv[d];
        const float4 hv = *(const float4*)(h + (long)s * D_H + lane * 4);
        float* ap = agg + (long)d * D_H + lane * 4;
        atomicAdd(ap + 0, hv.x * nrm);
        atomicAdd(ap + 1, hv.y * nrm);
        atomicAdd(ap + 2, hv.z * nrm);
        atomicAdd(ap + 3, hv.w * nrm);
    }
}

__global__ void bias_relu_kernel(const float* __restrict__ in,
                                 const float* __restrict__ b,
                                 float* __restrict__ out, int n) {
    long idx = (long)blockIdx.x * blockDim.x + threadIdx.x;
    if (idx >= (long)n * D_H) return;
    out[idx] = fmaxf(in[idx] + b[idx & (D_H - 1)], 0.f);
}

// ---------------------------------------------------------------------------
// Row-wise log_softmax over 32 classes: one wave per row, lane == class.
// In-place on d_out (each lane reads its own element before the write).
// ---------------------------------------------------------------------------
__global__ void logsoftmax_kernel(float* __restrict__ io, int n) {
    int w    = (int)(((long)blockIdx.x * blockDim.x + threadIdx.x) >> 5);
    int lane = threadIdx.x & 31;
    if (w >= n) return;
    float v = io[(long)w * D_OUT + lane];
    float m = v;
#pragma unroll
    for (int o = 16; o > 0; o >>= 1) m = fmaxf(m, __shfl_xor(m, o, 32));
    float s = __expf(v - m);
#pragma unroll
    for (int o = 16; o > 0; o >>= 1) s += __shfl_xor(s, o, 32);
    io[(long)w * D_OUT + lane] = v - m - __logf(s);
}

// ---------------------------------------------------------------------------
extern "C" void kernel_launch(void* const* d_in, const int* in_sizes, int n_in,
                              void* d_out, int out_size, void* d_ws, size_t ws_size,
                              hipStream_t stream) {
    const float* x    = (const float*)d_in[0];
    const int*   ei   = (const int*)d_in[1];
    const float* W1   = (const float*)d_in[2];
    const float* b1   = (const float*)d_in[3];
    const float* W2   = (const float*)d_in[4];
    const float* b2   = (const float*)d_in[5];
    const float* fw1  = (const float*)d_in[6];
    const float* fb1  = (const float*)d_in[7];
    const float* fw2  = (const float*)d_in[8];
    const float* fb2  = (const float*)d_in[9];
    float* out = (float*)d_out;

    const int n = in_sizes[0] / D_IN;     // 100000
    const int e = in_sizes[1] / 2;        // 3200000
    const int* src = ei;
    const int* dst = ei + e;

    // workspace layout (256B aligned)
    size_t off = 0;
    auto alloc = [&](size_t bytes) {
        void* p = (char*)d_ws + off;
        off += (bytes + 255) & ~(size_t)255;
        return p;
    };
    float* deg  = (float*)alloc((size_t)n * 4);
    float* dinv = (float*)alloc((size_t)n * 4);
    float* bufA = (float*)alloc((size_t)n * D_H * 4);
    float* bufB = (float*)alloc((size_t)n * D_H * 4);
    (void)ws_size; (void)n_in; (void)out_size;

    const int ew_blocks = ((long)n * D_H + 255) / 256;           // elementwise over n*128
    const int mtiles    = (n + 15) / 16;                          // 6250
    const int gemm_blk  = (mtiles + 7) / 8;                       // 782

    // degrees + symmetric normalization
    init_deg_kernel<<<(n + 255) / 256, 256, 0, stream>>>(deg, n);
    edge_deg_kernel<<<(e + 255) / 256, 256, 0, stream>>>(dst, deg, e);
    dinv_kernel<<<(n + 255) / 256, 256, 0, stream>>>(deg, dinv, n);

    // layer 1: h = x @ W1 ; agg = scatter ; relu(agg + b1)
    gemm_wmma_kernel<D_IN, D_H, 64, false, false>
        <<<gemm_blk, 256, 0, stream>>>(x, W1, nullptr, bufA, n);
    selfloop_kernel<<<ew_blocks, 256, 0, stream>>>(bufA, dinv, bufB, n);
    edge_scatter_kernel<<<4096, 256, 0, stream>>>(bufA, src, dst, dinv, bufB, e);
    bias_relu_kernel<<<ew_blocks, 256, 0, stream>>>(bufB, b1, bufA, n);

    // layer 2: h = h @ W2 ; agg = scatter ; relu(agg + b2)
    gemm_wmma_kernel<D_H, D_H, 64, false, false>
        <<<gemm_blk, 256, 0, stream>>>(bufA, W2, nullptr, bufB, n);
    selfloop_kernel<<<ew_blocks, 256, 0, stream>>>(bufB, dinv, bufA, n);
    edge_scatter_kernel<<<4096, 256, 0, stream>>>(bufB, src, dst, dinv, bufA, e);
    bias_relu_kernel<<<ew_blocks, 256, 0, stream>>>(bufA, b2, bufB, n);

    // MLP head: relu(h @ fw1 + fb1) ; h @ fw2 + fb2 (written straight to d_out)
    gemm_wmma_kernel<D_H, D_H, 64, true, true>
        <<<gemm_blk, 256, 0, stream>>>(bufB, fw1, fb1, bufA, n);
    gemm_wmma_kernel<D_H, D_OUT, 64, true, false>
        <<<gemm_blk, 256, 0, stream>>>(bufA, fw2, fb2, out, n);

    // log_softmax over 32 classes, one wave32 per row, in-place
    logsoftmax_kernel<<<((long)n * 32 + 255) / 256, 256, 0, stream>>>(out, n);
}